// GCN_simple_7687991460232
// MI455X (gfx1250) — compile-verified
//
#include <hip/hip_runtime.h>
#include <math.h>

#define N_NODES 50000
#define N_EDGES 400000
#define HEADS   2
#define DDIM    128
#define F_IN    5
#define HD      (HEADS*DDIM)   /* 256 */
#define BN_EPS  1e-5f

typedef __attribute__((ext_vector_type(2))) float v2f;
typedef __attribute__((ext_vector_type(8))) float v8f;

// ---- monotonic float<->uint encoding for atomicMax-based segment max ----
__device__ __forceinline__ unsigned enc_f32(float f) {
    unsigned u = __float_as_uint(f);
    return (u & 0x80000000u) ? ~u : (u | 0x80000000u);
}
__device__ __forceinline__ float dec_f32(unsigned u) {
    unsigned v = (u & 0x80000000u) ? (u & 0x7FFFFFFFu) : ~u;
    return __uint_as_float(v);
}
__device__ __forceinline__ void atomAddF(float* p, float v) {
    unsafeAtomicAdd(p, v);   // lowers to global_atomic_add_f32 on gfx1250
}

// ---------------- K0: init scratch (must run every call) ----------------
__global__ void k_init(float* __restrict__ den, float* __restrict__ cnt,
                       float* __restrict__ ssum, unsigned* __restrict__ mx) {
    long i = (long)blockIdx.x * blockDim.x + threadIdx.x;
    const long total = (long)N_NODES * HD;
    for (long j = i; j < total; j += (long)gridDim.x * blockDim.x) ssum[j] = 0.f;
    if (i < N_NODES) cnt[i] = 0.f;
    if (i < (long)N_NODES * HEADS) { den[i] = 0.f; mx[i] = 0u; } // 0 == enc(<-inf)
}

// ---------------- K1: BN1 + node linear transforms (K=5) ----------------
__global__ __launch_bounds__(256)
void k_node(const float* __restrict__ h,
            const float* __restrict__ gamma1, const float* __restrict__ beta1,
            const float* __restrict__ rm1, const float* __restrict__ rv1,
            const float* __restrict__ Wl, const float* __restrict__ bl,
            const float* __restrict__ Wr, const float* __restrict__ br,
            float* __restrict__ xl, float* __restrict__ xr) {
    int n = blockIdx.x;
    int c = threadIdx.x;           // 0..255 output column
    float hbn[F_IN];
#pragma unroll
    for (int k = 0; k < F_IN; ++k) {
        float v = h[n * F_IN + k];
        hbn[k] = (v - rm1[k]) * rsqrtf(rv1[k] + BN_EPS) * gamma1[k] + beta1[k];
    }
    float sl = bl[c], sr = br[c];
#pragma unroll
    for (int k = 0; k < F_IN; ++k) {
        sl += hbn[k] * Wl[k * HD + c];
        sr += hbn[k] * Wr[k * HD + c];
    }
    xl[(long)n * HD + c] = sl;
    xr[(long)n * HD + c] = sr;
}

// ---------------- K2: per-edge GATv2 logits + segment max + counts ------
// one wave32 per edge; lane covers 8 contiguous hd values (lanes 0-15 head0,
// lanes 16-31 head1); butterfly reduce within each 16-lane half.
__global__ __launch_bounds__(256)
void k_edge_logits(const int* __restrict__ ei, const float* __restrict__ ew,
                   const float* __restrict__ We, const float* __restrict__ att,
                   const float* __restrict__ xl, const float* __restrict__ xr,
                   float* __restrict__ logits, unsigned* __restrict__ mx,
                   float* __restrict__ cnt) {
    int wave = threadIdx.x >> 5, lane = threadIdx.x & 31;
    int e = blockIdx.x * 8 + wave;
    if (e >= N_EDGES) return;
    int src = ei[e];
    int dst = ei[N_EDGES + e];
    float w = ew[e];
    const float* pl = xl + (long)src * HD;
    const float* pr = xr + (long)dst * HD;
    float s = 0.f;
    int base = lane * 8;
#pragma unroll
    for (int j = 0; j < 8; ++j) {
        int hd = base + j;
        float z = pl[hd] + pr[hd] + w * We[hd];
        z = z > 0.f ? z : 0.2f * z;          // leaky_relu slope 0.2
        s += att[hd] * z;
    }
    s += __shfl_xor(s, 8, 32);
    s += __shfl_xor(s, 4, 32);
    s += __shfl_xor(s, 2, 32);
    s += __shfl_xor(s, 1, 32);
    if (lane == 0 || lane == 16) {
        int hh = lane >> 4;
        logits[(long)e * HEADS + hh] = s;
        atomicMax(&mx[(long)dst * HEADS + hh], enc_f32(s));
        if (lane == 0) atomAddF(&cnt[dst], 1.f);
    }
}

// ---------------- K3: exp(logit - max) + segment denominator ------------
__global__ __launch_bounds__(256)
void k_edge_exp(const int* __restrict__ ei, float* __restrict__ logits,
                const unsigned* __restrict__ mx, float* __restrict__ den) {
    long i = (long)blockIdx.x * blockDim.x + threadIdx.x;
    if (i >= (long)N_EDGES * HEADS) return;
    int e = (int)(i >> 1), hh = (int)(i & 1);
    int dst = ei[N_EDGES + e];
    float m = dec_f32(mx[(long)dst * HEADS + hh]);
    float ex = expf(logits[i] - m);
    logits[i] = ex;                    // reuse buffer for ex
    atomAddF(&den[(long)dst * HEADS + hh], ex);
}

// ---------------- K4: alpha-weighted scatter of messages ---------------
__global__ __launch_bounds__(256)
void k_edge_scatter(const int* __restrict__ ei, const float* __restrict__ ex,
                    const float* __restrict__ den, const float* __restrict__ xl,
                    float* __restrict__ ssum) {
    int wave = threadIdx.x >> 5, lane = threadIdx.x & 31;
    int e = blockIdx.x * 8 + wave;
    if (e >= N_EDGES) return;
    int src = ei[e], dst = ei[N_EDGES + e];
    int hh = lane >> 4;
    float alpha = ex[(long)e * HEADS + hh] / (den[(long)dst * HEADS + hh] + 1e-16f);
    const float* pl = xl + (long)src * HD;
    float* ps = ssum + (long)dst * HD;
    int base = lane * 8;
#pragma unroll
    for (int j = 0; j < 8; ++j) {
        int hd = base + j;
        atomAddF(&ps[hd], pl[hd] * alpha);     // ssum resident in 192MB L2
    }
}

// ---- K5: fused head-mean + fc1 (f32 WMMA) + BN2 + lrelu + fc2 ---------
__global__ __launch_bounds__(256)
void k_fc(const float* __restrict__ ssum, const float* __restrict__ cnt,
          const float* __restrict__ bias_gat,
          const float* __restrict__ W1, const float* __restrict__ b1,
          const float* __restrict__ gamma2, const float* __restrict__ beta2,
          const float* __restrict__ rm2, const float* __restrict__ rv2,
          const float* __restrict__ W2, const float* __restrict__ b2,
          float* __restrict__ out) {
    __shared__ float Y[16 * 128];
    __shared__ float Z[16 * 128];
    int row0 = blockIdx.x * 16;          // N = 50000 = 3125 * 16 exactly
    int t = threadIdx.x;

    // Stage Y tile: head-mean / count-mean / +bias_gat fused into the load
    {
        int r = t >> 4;
        int c0 = (t & 15) * 8;
        int n = row0 + r;
        float inv = 0.5f / fmaxf(cnt[n], 1.f);
        const float* p = ssum + (long)n * HD;
#pragma unroll
        for (int j = 0; j < 8; ++j) {
            int d = c0 + j;
            Y[r * 128 + d] = (p[d] + p[128 + d]) * inv + bias_gat[d];
        }
    }
    __syncthreads();

    // Each of 8 waves owns a 16x16 output tile; K=128 in steps of 4.
    int wave = t >> 5, lane = t & 31;
    int c0 = wave * 16;
    int laneN = lane & 15;
    int koff = (lane >> 4) * 2;          // lanes 16-31 carry K+2,K+3
    v8f acc = {0.f, 0.f, 0.f, 0.f, 0.f, 0.f, 0.f, 0.f};
    for (int k = 0; k < 128; k += 4) {
        v2f a, b;
        a.x = Y[laneN * 128 + k + koff];
        a.y = Y[laneN * 128 + k + koff + 1];
        int kk = k + koff;
        b.x = W1[kk * 128 + c0 + laneN];
        b.y = W1[(kk + 1) * 128 + c0 + laneN];
        acc = __builtin_amdgcn_wmma_f32_16x16x4_f32(
            false, a, false, b, (short)0, acc, false, false);
    }

    // Epilogue: +b1, BN2, leaky_relu(0.01); D layout M = v + 8*(lane>=16)
#pragma unroll
    for (int v = 0; v < 8; ++v) {
        int m = v + (lane >> 4) * 8;
        int col = c0 + laneN;
        float z = acc[v] + b1[col];
        z = (z - rm2[col]) * rsqrtf(rv2[col] + BN_EPS) * gamma2[col] + beta2[col];
        z = z > 0.f ? z : 0.01f * z;
        Z[m * 128 + col] = z;
    }
    __syncthreads();

    // fc2: 16 rows x 5 cols, K=128
    if (t < 16 * F_IN) {
        int r = t / F_IN, c = t % F_IN;
        float o = b2[c];
        const float* zr = &Z[r * 128];
#pragma unroll 8
        for (int d = 0; d < 128; ++d) o += zr[d] * W2[d * F_IN + c];
        out[(long)(row0 + r) * F_IN + c] = o;
    }
}

extern "C" void kernel_launch(void* const* d_in, const int* in_sizes, int n_in,
                              void* d_out, int out_size, void* d_ws, size_t ws_size,
                              hipStream_t stream) {
    const float* h        = (const float*)d_in[0];
    const int*   ei       = (const int*)  d_in[1];
    const float* ew       = (const float*)d_in[2];
    const float* gamma1   = (const float*)d_in[3];
    const float* beta1    = (const float*)d_in[4];
    const float* rm1      = (const float*)d_in[5];
    const float* rv1      = (const float*)d_in[6];
    const float* Wl       = (const float*)d_in[7];
    const float* bl       = (const float*)d_in[8];
    const float* Wr       = (const float*)d_in[9];
    const float* br       = (const float*)d_in[10];
    const float* We       = (const float*)d_in[11];
    const float* att      = (const float*)d_in[12];
    const float* bias_gat = (const float*)d_in[13];
    const float* W1       = (const float*)d_in[14];
    const float* b1       = (const float*)d_in[15];
    const float* gamma2   = (const float*)d_in[16];
    const float* beta2    = (const float*)d_in[17];
    const float* rm2      = (const float*)d_in[18];
    const float* rv2      = (const float*)d_in[19];
    const float* W2       = (const float*)d_in[20];
    const float* b2       = (const float*)d_in[21];
    float* out = (float*)d_out;

    float* ws = (float*)d_ws;
    size_t o = 0;
    float* xl     = ws + o; o += (size_t)N_NODES * HD;      // 12.8M
    float* xr     = ws + o; o += (size_t)N_NODES * HD;      // 12.8M
    float* logits = ws + o; o += (size_t)N_EDGES * HEADS;   // 0.8M (reused as ex)
    float* den    = ws + o; o += (size_t)N_NODES * HEADS;
    float* cnt    = ws + o; o += (size_t)N_NODES;
    float* ssum   = ws + o; o += (size_t)N_NODES * HD;      // 12.8M
    unsigned* mx  = (unsigned*)(ws + o); o += (size_t)N_NODES * HEADS;

    const int T = 256;
    k_init<<<(N_NODES * HD + T - 1) / T, T, 0, stream>>>(den, cnt, ssum, mx);
    k_node<<<N_NODES, T, 0, stream>>>(h, gamma1, beta1, rm1, rv1,
                                      Wl, bl, Wr, br, xl, xr);
    k_edge_logits<<<N_EDGES / 8, T, 0, stream>>>(ei, ew, We, att, xl, xr,
                                                 logits, mx, cnt);
    k_edge_exp<<<(N_EDGES * HEADS + T - 1) / T, T, 0, stream>>>(ei, logits, mx, den);
    k_edge_scatter<<<N_EDGES / 8, T, 0, stream>>>(ei, logits, den, xl, ssum);
    k_fc<<<N_NODES / 16, T, 0, stream>>>(ssum, cnt, bias_gat, W1, b1,
                                         gamma2, beta2, rm2, rv2, W2, b2, out);
}